// EMDModule_19645180412396
// MI455X (gfx1250) — compile-verified
//
#include <hip/hip_runtime.h>

typedef __attribute__((ext_vector_type(2))) float v2f;
typedef __attribute__((ext_vector_type(8))) float v8f;

#define NPTS_MAX 2048
#define BLOCK 256
#define ROWS_PER_BLOCK 128   // 8 waves * 16 rows each

// ---------------------------------------------------------------------------
// Kernel A: per level, compute rowscale[i] = remainL[i] / (1e-9 + sum_j w_pre)
// where w_pre = exp(level*d2[i][j]) * remainR[j].
// Wave owns 16 rows of xyz1; loops 16-wide tiles of xyz2 (staged in LDS).
// WMMA computes |p|^2 - 2p.q with C = inline 0; |q|^2 is folded into the
// exp argument with a single FMA (saves the 8-mov C-splat per tile).
// ---------------------------------------------------------------------------
__global__ void __launch_bounds__(BLOCK)
emd_rowsum_kernel(const float* __restrict__ xyz1, const float* __restrict__ xyz2,
                  const float* __restrict__ remainL, const float* __restrict__ remainR,
                  float* __restrict__ rowscale, float lscale, int n, int m)
{
    __shared__ float4 s_pts[NPTS_MAX];
    __shared__ float  s_fac[NPTS_MAX];
    const int b   = blockIdx.y;
    const int tid = threadIdx.x;

    for (int j = tid; j < m; j += BLOCK) {
        const float x = xyz2[(b * m + j) * 3 + 0];
        const float y = xyz2[(b * m + j) * 3 + 1];
        const float z = xyz2[(b * m + j) * 3 + 2];
        s_pts[j] = make_float4(x, y, z, x * x + y * y + z * z);
        s_fac[j] = remainR[b * m + j];
    }
    __syncthreads();

    const int  wave = tid >> 5;
    const int  lane = tid & 31;
    const bool hi   = lane >= 16;
    const int  i0   = blockIdx.x * ROWS_PER_BLOCK + wave * 16;
    const int  row  = i0 + (lane & 15);

    const float px = xyz1[(b * n + row) * 3 + 0];
    const float py = xyz1[(b * n + row) * 3 + 1];
    const float pz = xyz1[(b * n + row) * 3 + 2];
    const float np = px * px + py * py + pz * pz;
    v2f a;                               // A: 16x4, rows=i, K={-2x,-2y,-2z,|p|^2}
    a.x = hi ? (-2.0f * pz) : (-2.0f * px);   // K=2 | K=0
    a.y = hi ? np            : (-2.0f * py);  // K=3 | K=1

    float acc[8];
#pragma unroll
    for (int v = 0; v < 8; ++v) acc[v] = 0.0f;

    const int ntiles = m >> 4;
    for (int jt = 0; jt < ntiles; ++jt) {
        const int    jj = (jt << 4) + (lane & 15);
        const float4 f  = s_pts[jj];
        const float  rr = s_fac[jj];
        v2f bb;                          // B: 4x16, K rows {x,y,z,1}
        bb.x = hi ? f.z : f.x;           // K=2 | K=0
        bb.y = hi ? 1.0f : f.y;          // K=3 | K=1
        v8f c = {};                      // SRC2 = inline 0
        v8f d = __builtin_amdgcn_wmma_f32_16x16x4_f32(false, a, false, bb,
                                                      (short)0, c, false, false);
        const float fnq = lscale * f.w;  // fold |q|^2 into the exp argument
#pragma unroll
        for (int v = 0; v < 8; ++v) {
            const float e = __builtin_amdgcn_exp2f(fmaf(lscale, d[v], fnq));
            acc[v] += e * rr;
        }
    }
    // sum over the 16 lanes of each half-wave (N dimension of the tile)
#pragma unroll
    for (int v = 0; v < 8; ++v) {
#pragma unroll
        for (int off = 1; off < 16; off <<= 1)
            acc[v] += __shfl_xor(acc[v], off, 32);
    }
    if ((lane & 15) == 0) {
        const int base = i0 + (hi ? 8 : 0);
#pragma unroll
        for (int v = 0; v < 8; ++v) {
            const int   r  = base + v;
            const float rl = remainL[b * n + r];
            rowscale[b * n + r] = rl / (1e-9f + acc[v]);
        }
    }
}

// ---------------------------------------------------------------------------
// Kernel B: column sums s_col[j] = remainR[j] * sum_i exp(level*d2)*rowscale[i];
// then r = min(remainR/(1e-9+s_col),1), q[j] = remainR[j]*r,
// remainR[j] = max(remainR[j] - r*s_col, 0).  Wave owns 16 columns (transposed
// tiles: A built from xyz2, B/C from xyz1 in LDS).  Same C=0 + FMA-fold trick.
// ---------------------------------------------------------------------------
__global__ void __launch_bounds__(BLOCK)
emd_colsum_kernel(const float* __restrict__ xyz1, const float* __restrict__ xyz2,
                  const float* __restrict__ rowscale, float* __restrict__ remainR,
                  float* __restrict__ qbuf, float lscale, int n, int m)
{
    __shared__ float4 s_pts[NPTS_MAX];
    __shared__ float  s_fac[NPTS_MAX];
    const int b   = blockIdx.y;
    const int tid = threadIdx.x;

    for (int i = tid; i < n; i += BLOCK) {
        const float x = xyz1[(b * n + i) * 3 + 0];
        const float y = xyz1[(b * n + i) * 3 + 1];
        const float z = xyz1[(b * n + i) * 3 + 2];
        s_pts[i] = make_float4(x, y, z, x * x + y * y + z * z);
        s_fac[i] = rowscale[b * n + i];
    }
    __syncthreads();

    const int  wave = tid >> 5;
    const int  lane = tid & 31;
    const bool hi   = lane >= 16;
    const int  j0   = blockIdx.x * ROWS_PER_BLOCK + wave * 16;
    const int  col  = j0 + (lane & 15);

    const float qx = xyz2[(b * m + col) * 3 + 0];
    const float qy = xyz2[(b * m + col) * 3 + 1];
    const float qz = xyz2[(b * m + col) * 3 + 2];
    const float nq = qx * qx + qy * qy + qz * qz;
    v2f a;
    a.x = hi ? (-2.0f * qz) : (-2.0f * qx);
    a.y = hi ? nq            : (-2.0f * qy);

    float acc[8];
#pragma unroll
    for (int v = 0; v < 8; ++v) acc[v] = 0.0f;

    const int ntiles = n >> 4;
    for (int it = 0; it < ntiles; ++it) {
        const int    ii = (it << 4) + (lane & 15);
        const float4 f  = s_pts[ii];
        const float  rs = s_fac[ii];
        v2f bb;
        bb.x = hi ? f.z : f.x;
        bb.y = hi ? 1.0f : f.y;
        v8f c = {};
        v8f d = __builtin_amdgcn_wmma_f32_16x16x4_f32(false, a, false, bb,
                                                      (short)0, c, false, false);
        const float fnp = lscale * f.w;
#pragma unroll
        for (int v = 0; v < 8; ++v) {
            const float e = __builtin_amdgcn_exp2f(fmaf(lscale, d[v], fnp));
            acc[v] += e * rs;
        }
    }
#pragma unroll
    for (int v = 0; v < 8; ++v) {
#pragma unroll
        for (int off = 1; off < 16; off <<= 1)
            acc[v] += __shfl_xor(acc[v], off, 32);
    }
    if ((lane & 15) == 0) {
        const int base = j0 + (hi ? 8 : 0);
#pragma unroll
        for (int v = 0; v < 8; ++v) {
            const int   jj   = base + v;
            const float rr   = remainR[b * m + jj];
            const float scol = rr * acc[v];
            const float r    = fminf(rr / (1e-9f + scol), 1.0f);
            qbuf[b * m + jj]    = rr * r;
            remainR[b * m + jj] = fmaxf(rr - r * scol, 0.0f);
        }
    }
}

// ---------------------------------------------------------------------------
// Kernel C: final weights w = exp(level*d2)*q[j]*rowscale[i].
// Accumulate cost += w*sqrt(max(d2,1e-30)) into out[b] (atomic) and
// remainL[i] = max(remainL[i] - sum_j w, 0).  Keeps C = |q|^2 in the WMMA
// because the true d2 is needed for the sqrt.
// ---------------------------------------------------------------------------
__global__ void __launch_bounds__(BLOCK)
emd_cost_kernel(const float* __restrict__ xyz1, const float* __restrict__ xyz2,
                const float* __restrict__ rowscale, float* __restrict__ remainL,
                const float* __restrict__ qbuf, float* __restrict__ out,
                float lscale, int n, int m)
{
    __shared__ float4 s_pts[NPTS_MAX];
    __shared__ float  s_fac[NPTS_MAX];
    const int b   = blockIdx.y;
    const int tid = threadIdx.x;

    for (int j = tid; j < m; j += BLOCK) {
        const float x = xyz2[(b * m + j) * 3 + 0];
        const float y = xyz2[(b * m + j) * 3 + 1];
        const float z = xyz2[(b * m + j) * 3 + 2];
        s_pts[j] = make_float4(x, y, z, x * x + y * y + z * z);
        s_fac[j] = qbuf[b * m + j];
    }
    __syncthreads();

    const int  wave = tid >> 5;
    const int  lane = tid & 31;
    const bool hi   = lane >= 16;
    const int  i0   = blockIdx.x * ROWS_PER_BLOCK + wave * 16;
    const int  row  = i0 + (lane & 15);

    const float px = xyz1[(b * n + row) * 3 + 0];
    const float py = xyz1[(b * n + row) * 3 + 1];
    const float pz = xyz1[(b * n + row) * 3 + 2];
    const float np = px * px + py * py + pz * pz;
    v2f a;
    a.x = hi ? (-2.0f * pz) : (-2.0f * px);
    a.y = hi ? np            : (-2.0f * py);

    float accW[8], accC[8];
#pragma unroll
    for (int v = 0; v < 8; ++v) { accW[v] = 0.0f; accC[v] = 0.0f; }

    const int ntiles = m >> 4;
    for (int jt = 0; jt < ntiles; ++jt) {
        const int    jj = (jt << 4) + (lane & 15);
        const float4 f  = s_pts[jj];
        const float  qv = s_fac[jj];
        v2f bb;
        bb.x = hi ? f.z : f.x;
        bb.y = hi ? 1.0f : f.y;
        v8f c;
#pragma unroll
        for (int v = 0; v < 8; ++v) c[v] = f.w;   // + |q|^2 -> true d2
        v8f d = __builtin_amdgcn_wmma_f32_16x16x4_f32(false, a, false, bb,
                                                      (short)0, c, false, false);
#pragma unroll
        for (int v = 0; v < 8; ++v) {
            const float e  = __builtin_amdgcn_exp2f(lscale * d[v]);
            const float wq = e * qv;
            accW[v] += wq;
            accC[v] += wq * __builtin_amdgcn_sqrtf(fmaxf(d[v], 1e-30f));
        }
    }
#pragma unroll
    for (int v = 0; v < 8; ++v) {
#pragma unroll
        for (int off = 1; off < 16; off <<= 1) {
            accW[v] += __shfl_xor(accW[v], off, 32);
            accC[v] += __shfl_xor(accC[v], off, 32);
        }
    }
    if ((lane & 15) == 0) {
        const int base = i0 + (hi ? 8 : 0);
        float cl = 0.0f;
#pragma unroll
        for (int v = 0; v < 8; ++v) {
            const int   r  = base + v;
            const float rs = rowscale[b * n + r];
            const float rl = remainL[b * n + r];
            remainL[b * n + r] = fmaxf(rl - rs * accW[v], 0.0f);
            cl += rs * accC[v];
        }
        atomicAdd(&out[b], cl);
    }
}

// ---------------------------------------------------------------------------
// Init: remainL = multiL, remainR = multiR, out = 0 (fresh every call).
// ---------------------------------------------------------------------------
__global__ void emd_init_kernel(float* remainL, float* remainR, float* out,
                                float multiL, float multiR, int bn, int bm, int b)
{
    const int t = blockIdx.x * blockDim.x + threadIdx.x;
    if (t < bn) remainL[t] = multiL;
    if (t < bm) remainR[t] = multiR;
    if (t < b)  out[t] = 0.0f;
}

extern "C" void kernel_launch(void* const* d_in, const int* in_sizes, int n_in,
                              void* d_out, int out_size, void* d_ws, size_t ws_size,
                              hipStream_t stream)
{
    const float* xyz1 = (const float*)d_in[0];
    const float* xyz2 = (const float*)d_in[1];
    float*       out  = (float*)d_out;

    const int b = out_size;                 // 8
    const int n = in_sizes[0] / (3 * b);    // 2048
    const int m = in_sizes[1] / (3 * b);    // 2048

    float* remainL  = (float*)d_ws;
    float* remainR  = remainL + (size_t)b * n;
    float* rowscale = remainR + (size_t)b * m;
    float* qbuf     = rowscale + (size_t)b * n;

    const float multiL = (float)((m / n) > 1 ? (m / n) : 1);
    const float multiR = (float)((n / m) > 1 ? (n / m) : 1);

    const int tot = b * (n > m ? n : m);
    emd_init_kernel<<<(tot + 255) / 256, 256, 0, stream>>>(remainL, remainR, out,
                                                           multiL, multiR,
                                                           b * n, b * m, b);

    // levels: -4^j for j = 7..-1, then 0; prefold log2(e) for v_exp_f32.
    const float L2E = 1.4426950408889634f;
    const float levels[10] = {-16384.f, -4096.f, -1024.f, -256.f, -64.f,
                              -16.f, -4.f, -1.f, -0.25f, 0.f};

    dim3 gridRows(n / ROWS_PER_BLOCK, b);
    dim3 gridCols(m / ROWS_PER_BLOCK, b);

    for (int l = 0; l < 10; ++l) {
        const float ls = levels[l] * L2E;
        emd_rowsum_kernel<<<gridRows, BLOCK, 0, stream>>>(xyz1, xyz2, remainL,
                                                          remainR, rowscale, ls, n, m);
        emd_colsum_kernel<<<gridCols, BLOCK, 0, stream>>>(xyz1, xyz2, rowscale,
                                                          remainR, qbuf, ls, n, m);
        emd_cost_kernel<<<gridRows, BLOCK, 0, stream>>>(xyz1, xyz2, rowscale,
                                                        remainL, qbuf, out, ls, n, m);
    }
}